// Ladder2D_77129022701988
// MI455X (gfx1250) — compile-verified
//
#include <hip/hip_runtime.h>
#include <hip/hip_bf16.h>

// Ladder2D: y[n,co,h,w] = -sum_{cin,kh,kw} | x[n,cin,h+kh-1,w+kw-1] - pat[co,cin,kh*3+kw] |
// x: (8,64,32,32) f32, pat: (128,64,9) f32, y: (8,128,32,32) f32. pad=1 zero-padding.
//
// Tiling: 256 WGs = (n: 8) x (row-pair hp: 16) x (cout-half ch: 2).
// Each WG: 64 couts x 2 rows x 32 cols. Each thread: 2 couts x 2 rows micro-tile.
// pat half (144 KB) resident in LDS via TDM; x halo (4 rows, 34 KB) via async-to-LDS.

#define NIMG   8
#define CIN    64
#define HH     32
#define WW     32
#define COUT   128
#define KK     9
#define CPK    (CIN * KK)        // 576
#define XPITCH 34                // cols -1..32 (zero borders at 0 and 33)
#define XROWS  4                 // halo rows h0-1 .. h0+2
#define PATH   (64 * CPK)        // 36864 f32 = 144 KB (half of pat, 64 couts)
#define SLICE  (PATH / 8)        // 4608 f32 = 18 KB per wave (one TDM each)
#define NGRP   4                 // 4 groups of 16 couts in the compute loop

typedef __attribute__((ext_vector_type(4))) unsigned int u32x4;
typedef __attribute__((ext_vector_type(8))) unsigned int u32x8;

__global__ __launch_bounds__(256)
void ladder2d_l1_kernel(const float* __restrict__ x,
                        const float* __restrict__ pat,
                        float* __restrict__ out) {
  __shared__ float s_x[CIN * XROWS * XPITCH];  // 34816 B  (x halo rows, zero-padded)
  __shared__ float s_pat[PATH];                // 147456 B (64-cout half of pat)
  // total 182272 B = 178 KB <= 320 KB CDNA5 LDS/workgroup

  const int tid = threadIdx.x;              // 0..255 (8 waves)
  const int wv  = tid >> 5;                 // wave id 0..7 (uniform per wave)
  const int n   = blockIdx.x >> 5;          // image
  const int hp  = (blockIdx.x >> 1) & 15;   // row pair -> rows 2hp, 2hp+1
  const int ch  = blockIdx.x & 1;           // cout half (0: couts 0..63, 1: 64..127)
  const int h0  = 2 * hp;
  const int hbase = h0 - 1;                 // first halo row

  // ---- zero-fill s_x: covers pad columns (0,33) and out-of-range halo rows ----
  for (int i = tid; i < CIN * XROWS * XPITCH; i += 256) s_x[i] = 0.0f;
  __syncthreads();  // zero stores must land before async writes to interior

  // ---- TDM: each wave DMAs its contiguous 1/8 slice of this WG's pat half ----
  {
    const uint64_t ga = (uint64_t)(uintptr_t)(pat + (size_t)ch * PATH + (size_t)wv * SLICE);
    u32x4 g0;
    g0[0] = 1u;                                               // count=1, user descriptor
    g0[1] = (uint32_t)(uintptr_t)(&s_pat[0])
            + (uint32_t)(wv * SLICE * 4);                     // lds_addr (bytes)
    g0[2] = (uint32_t)ga;                                     // global_addr[31:0]
    g0[3] = ((uint32_t)(ga >> 32) & 0x01FFFFFFu)              // global_addr[56:32]
            | 0x80000000u;                                    // type=2 @127:126

    u32x8 g1;
    g1[0] = 0x00020000u;                  // wg_mask=0, data_size=2 (4 bytes)
    g1[1] = ((uint32_t)SLICE) << 16;      // tensor_dim0[15:0]  @63:48
    g1[2] = (((uint32_t)SLICE) >> 16)     // tensor_dim0[31:16] @79:64
            | (1u << 16);                 // tensor_dim1=1 (lo16) @95:80
    g1[3] = ((uint32_t)SLICE) << 16;      // tensor_dim1 hi=0 | tile_dim0 @127:112
    g1[4] = 1u;                           // tile_dim1=1, tile_dim2=0 (unused)
    g1[5] = (uint32_t)SLICE;              // tensor_dim0_stride[31:0]
    g1[6] = 0u;                           // stride0 hi | stride1 lo
    g1[7] = 0u;                           // stride1 hi

    asm volatile("tensor_load_to_lds %0, %1" :: "s"(g0), "s"(g1) : "memory");
  }

  // ---- async-load interior of the 4 halo rows: element (cin, r, w) ----
  {
    const uint32_t lds_base = (uint32_t)(uintptr_t)(&s_x[0]);
    for (int e = tid; e < CIN * XROWS * WW; e += 256) {      // 8192 elems, 32/thread
      const int w   = e & 31;
      const int r   = (e >> 5) & 3;
      const int cin = e >> 7;
      const int hr  = hbase + r;
      if (hr >= 0 && hr < HH) {
        const uint32_t lds_off =
            lds_base + (uint32_t)(((cin * XROWS + r) * XPITCH + 1 + w) * 4);
        const uint64_t gaddr =
            (uint64_t)(uintptr_t)(x + (((n * CIN + cin) * HH + hr) * WW + w));
        asm volatile("global_load_async_to_lds_b32 %0, %1, off"
                     :: "v"(lds_off), "v"(gaddr) : "memory");
      }
    }
  }

  asm volatile("s_wait_asynccnt 0" ::: "memory");  // this wave's halo writes landed
  __builtin_amdgcn_s_wait_tensorcnt(0);            // this wave's pat slice landed
  __syncthreads();                                 // everything visible to all waves
  // ---- from here on: zero barriers, zero DMA waits, pure LDS + VALU ----

  const int w  = tid & 31;     // pixel column (lane id -> conflict-free LDS reads)
  const int cp = tid >> 5;     // cout-pair selector; uniform per wave -> LDS broadcast

  for (int g = 0; g < NGRP; ++g) {
    const float* __restrict__ pr0 = &s_pat[(g * 16 + 2 * cp + 0) * CPK];
    const float* __restrict__ pr1 = pr0 + CPK;

    // 2 couts x 2 rows -> 4 independent FP accumulation chains (VOPD-friendly)
    float acc00 = 0.0f, acc01 = 0.0f, acc10 = 0.0f, acc11 = 0.0f;

#pragma unroll 2
    for (int cin = 0; cin < CIN; ++cin) {
      const float* __restrict__ xr = &s_x[cin * XROWS * XPITCH + w];
      const float* __restrict__ p0 = pr0 + cin * KK;
      const float* __restrict__ p1 = pr1 + cin * KK;

      // 12 x values feed both output rows (row h0 uses r=0..2, row h0+1 uses r=1..3)
      float xv[XROWS][3];
#pragma unroll
      for (int r = 0; r < XROWS; ++r)
#pragma unroll
        for (int kw = 0; kw < 3; ++kw)
          xv[r][kw] = xr[r * XPITCH + kw];     // (1 + (w+kw-1)) column

#pragma unroll
      for (int kh = 0; kh < 3; ++kh) {
#pragma unroll
        for (int kw = 0; kw < 3; ++kw) {
          const int   i  = kh * 3 + kw;
          const float q0 = p0[i];
          const float q1 = p1[i];
          acc00 += fabsf(xv[kh][kw]     - q0);   // (cout,   h0)
          acc01 += fabsf(xv[kh][kw]     - q1);   // (cout+1, h0)
          acc10 += fabsf(xv[kh + 1][kw] - q0);   // (cout,   h0+1)
          acc11 += fabsf(xv[kh + 1][kw] - q1);   // (cout+1, h0+1)
        }
      }
    }

    const int cout = ch * 64 + g * 16 + 2 * cp;
    float* op = out + (((size_t)n * COUT + cout) * HH + h0) * WW + w;
    op[0]           = -acc00;   // (cout,   h0)
    op[WW]          = -acc10;   // (cout,   h0+1)
    op[HH * WW]     = -acc01;   // (cout+1, h0)
    op[HH * WW + WW] = -acc11;  // (cout+1, h0+1)
  }
}

extern "C" void kernel_launch(void* const* d_in, const int* in_sizes, int n_in,
                              void* d_out, int out_size, void* d_ws, size_t ws_size,
                              hipStream_t stream) {
  const float* x   = (const float*)d_in[0];   // 8*64*32*32
  const float* pat = (const float*)d_in[1];   // 128*64*9
  float* out = (float*)d_out;                 // 8*128*32*32
  (void)in_sizes; (void)n_in; (void)out_size; (void)d_ws; (void)ws_size;

  dim3 grid(NIMG * 16 * 2);   // 256 WGs: (n, row-pair, cout-half)
  dim3 block(256);            // 8 waves
  ladder2d_l1_kernel<<<grid, block, 0, stream>>>(x, pat, out);
}